// OverlapLoss_88502096101462
// MI455X (gfx1250) — compile-verified
//
#include <hip/hip_runtime.h>

// OverlapLoss: all-pairs IoU-overlap + repulsion loss on N=8192 boxes.
// Strategy: full NxN sum (symmetric) minus diagonal, halved. LDS-tiled
// 256x256 pair blocks; fp32 VALU hot loop; V_WMMA_F32_16X16X4_F32 used as an
// exact fp32 wave-wide reduction primitive. Fully deterministic (no atomics).

#define TILE 256

static constexpr float kEps = 1e-6f;
static constexpr float kRepelMargin = 0.06f;

typedef __attribute__((ext_vector_type(2))) float v2f;
typedef __attribute__((ext_vector_type(8))) float v8f;

// Sum p across the 32 lanes of a wave using V_WMMA_F32_16X16X4_F32.
// A = 16x4 all-ones. B = 4x16 with this wave's partials in its first VGPR and
// zeros in the second. D[m][n] = sum_k B[k][n]: the 16 column sums partition
// the 32 partials exactly once (layout-agnostic: every (k,n) slot is one of
// our 64 register values). Row M=0 lands in lanes 0..15 of d[0] (row M=8
// duplicates it in lanes 16..31), so 4 xor-shuffles over the 16-lane group
// complete the reduction. All fp32, deterministic.
__device__ __forceinline__ float wave_sum32(float p) {
  v2f a = {1.0f, 1.0f};
  v2f b = {p, 0.0f};
  v8f c = {};
  c = __builtin_amdgcn_wmma_f32_16x16x4_f32(false, a, false, b, (short)0, c,
                                            false, false);
  float r = c[0];
  r += __shfl_xor(r, 1, 32);
  r += __shfl_xor(r, 2, 32);
  r += __shfl_xor(r, 4, 32);
  r += __shfl_xor(r, 8, 32);
  return r;  // wave total, identical in every lane
}

// Per-pair terms (IOU_MARGIN = 0 so iou_pen == iou since iou >= 0).
__device__ __forceinline__ void pair_terms(
    float li, float ri, float bi, float ti, float xi, float yi, float ai, float vi,
    float lj, float rj, float bj, float tj, float xj, float yj, float aj, float vj,
    float& sI, float& sR) {
  float iw = fmaxf(fminf(ri, rj) - fmaxf(li, lj), 0.0f);
  float ih = fmaxf(fminf(ti, tj) - fmaxf(bi, bj), 0.0f);
  float inter = iw * ih;
  float uni = ai + aj - inter;
  float iou = inter / (uni + kEps);
  float vv = vi * vj;
  sI += iou * vv;
  float dx = xi - xj;
  float dy = yi - yj;
  float dist = sqrtf(dx * dx + dy * dy + kEps);
  float rp = fmaxf(kRepelMargin - dist, 0.0f);
  sR += rp * rp * vv;
}

__global__ void overlap_pair_kernel(const float* __restrict__ pos,
                                    const float* __restrict__ siz,
                                    const float* __restrict__ vm,
                                    float* __restrict__ ws, int n, int nblk) {
  __shared__ float4 tBox[TILE];  // (l, r, b, t) for the j tile
  __shared__ float4 tAux[TILE];  // (x, y, area, valid) for the j tile
  __shared__ float wpart[2][8];  // per-wave partials (8 waves of 32)

  const int tid = threadIdx.x;
  const int i = blockIdx.x * TILE + tid;
  const int j = blockIdx.y * TILE + tid;

  // Stage j tile into LDS (derived box quantities computed once per block).
  {
    float xj = pos[2 * j], yj = pos[2 * j + 1];
    float wj = siz[2 * j], hj = siz[2 * j + 1];
    float4 bx;
    bx.x = xj - 0.5f * wj;  // l
    bx.y = xj + 0.5f * wj;  // r
    bx.z = yj - 0.5f * hj;  // b
    bx.w = yj + 0.5f * hj;  // t
    float4 ax;
    ax.x = xj;
    ax.y = yj;
    ax.z = wj * hj;
    ax.w = vm[j];
    tBox[tid] = bx;
    tAux[tid] = ax;
  }
  __syncthreads();

  // This thread's i-box in registers.
  float xi = pos[2 * i], yi = pos[2 * i + 1];
  float wi = siz[2 * i], hi = siz[2 * i + 1];
  float vi = vm[i];
  float li = xi - 0.5f * wi, ri = xi + 0.5f * wi;
  float bi = yi - 0.5f * hi, ti = yi + 0.5f * hi;
  float ai = wi * hi;

  float sI = 0.0f, sR = 0.0f;
#pragma unroll 4
  for (int jj = 0; jj < TILE; ++jj) {
    float4 bx = tBox[jj];  // ds_load_b128 broadcast
    float4 ax = tAux[jj];
    pair_terms(li, ri, bi, ti, xi, yi, ai, vi,
               bx.x, bx.y, bx.z, bx.w, ax.x, ax.y, ax.z, ax.w, sI, sR);
  }

  // Wave-level reduction via WMMA, then fixed-order cross-wave combine.
  float wI = wave_sum32(sI);
  float wR = wave_sum32(sR);
  const int lane = tid & 31;
  const int wid = tid >> 5;
  if (lane == 0) {
    wpart[0][wid] = wI;
    wpart[1][wid] = wR;
  }
  __syncthreads();
  if (tid == 0) {
    float aI = 0.0f, aR = 0.0f;
#pragma unroll
    for (int k = 0; k < 8; ++k) {
      aI += wpart[0][k];
      aR += wpart[1][k];
    }
    const int bid = blockIdx.y * gridDim.x + blockIdx.x;
    ws[bid] = aI;          // full-matrix IoU partial for this 256x256 block
    ws[nblk + bid] = aR;   // full-matrix repel partial
  }
}

__global__ void overlap_finalize_kernel(const float* __restrict__ pos,
                                        const float* __restrict__ siz,
                                        const float* __restrict__ vm,
                                        const float* __restrict__ ws,
                                        float* __restrict__ out, int n,
                                        int nblk) {
  __shared__ float acc[6][8];
  const int tid = threadIdx.x;

  // 0: full IoU sum, 1: full repel sum, 2: diag IoU, 3: diag repel,
  // 4: sum(v), 5: sum(v^2)
  float s[6] = {0.f, 0.f, 0.f, 0.f, 0.f, 0.f};

  for (int b = tid; b < nblk; b += blockDim.x) {
    s[0] += ws[b];
    s[1] += ws[nblk + b];
  }
  for (int i = tid; i < n; i += blockDim.x) {
    float x = pos[2 * i], y = pos[2 * i + 1];
    float w = siz[2 * i], h = siz[2 * i + 1];
    float v = vm[i];
    float l = x - 0.5f * w, r = x + 0.5f * w;
    float bb = y - 0.5f * h, t = y + 0.5f * h;
    float a = w * h;
    pair_terms(l, r, bb, t, x, y, a, v, l, r, bb, t, x, y, a, v, s[2], s[3]);
    s[4] += v;
    s[5] += v * v;
  }

  const int lane = tid & 31;
  const int wid = tid >> 5;
#pragma unroll
  for (int q = 0; q < 6; ++q) {
    float t = wave_sum32(s[q]);
    if (lane == 0) acc[q][wid] = t;
  }
  __syncthreads();

  if (tid == 0) {
    float tot[6];
#pragma unroll
    for (int q = 0; q < 6; ++q) {
      float a = 0.0f;
#pragma unroll
      for (int k = 0; k < 8; ++k) a += acc[q][k];
      tot[q] = a;
    }
    float denom = 0.5f * (tot[4] * tot[4] - tot[5]) + kEps;  // #upper pairs + eps
    float sOv = 0.5f * (tot[0] - tot[2]);   // (full - diag) / 2
    float sRep = 0.5f * (tot[1] - tot[3]);
    // WEIGHT * RAMP * (overlap_raw + repel_raw), WEIGHT=0.5, RAMP=1
    out[0] = 0.5f * ((sOv + sRep) / denom);
  }
}

extern "C" void kernel_launch(void* const* d_in, const int* in_sizes, int n_in,
                              void* d_out, int out_size, void* d_ws,
                              size_t ws_size, hipStream_t stream) {
  const float* pos = (const float*)d_in[0];  // (N,2)
  const float* siz = (const float*)d_in[1];  // (N,2)
  const float* vm = (const float*)d_in[2];   // (N,)
  float* out = (float*)d_out;
  float* wsf = (float*)d_ws;

  const int n = in_sizes[2];    // N (8192)
  const int nc = n / TILE;      // 32 chunks per axis
  const int nblk = nc * nc;     // 1024 pair blocks; ws uses 2*nblk floats (8KB)

  dim3 grid(nc, nc);
  overlap_pair_kernel<<<grid, TILE, 0, stream>>>(pos, siz, vm, wsf, n, nblk);
  overlap_finalize_kernel<<<1, TILE, 0, stream>>>(pos, siz, vm, wsf, out, n,
                                                  nblk);
}